// HierarchicalLevel_4346506903770
// MI455X (gfx1250) — compile-verified
//
#include <hip/hip_runtime.h>
#include <math.h>

// GAT forward for MI455X (gfx1250, wave32).
// Roofline: GEMM = 1.64 GFLOP (trivial); edge scatter pass moves ~1.7 GB of
// random gather+atomic traffic -> memory bound. So: fp32 WMMA (16x16x4) for
// exact-precision projection; async copy engine (ASYNCcnt) for LDS staging;
// bandwidth-shaped wave-per-edge scatter for the heavy pass.

#define HEADS 4
#define CDIM 64
#define INDIM 64
#define HC 256           // HEADS*CDIM
#define NEG_SLOPE 0.2f
#define LN_EPS 1e-5f

typedef float v2f __attribute__((ext_vector_type(2)));
typedef float v8f __attribute__((ext_vector_type(8)));

// ---------------------------------------------------------------------------
// Kernel 1: xp = x @ W^T  via V_WMMA_F32_16X16X4_F32 (full fp32 precision).
// x: [N,64] row-major, W: [256,64] row-major, xp: [N,256] row-major.
// Block = 256 threads = 8 waves; wave w computes rows [blk*128 + w*16, +16)
// across all 256 output columns (16 column tiles x 16 K-blocks of 4).
// x tile staged into LDS with the CDNA5 async copy engine (no VGPR bounce).
// ---------------------------------------------------------------------------
#define LDS_STRIDE 68    // 68 floats = 272 B: 16B-aligned rows, conflict-free frags

__global__ __launch_bounds__(256) void gat_gemm_wmma(
    const float* __restrict__ x, const float* __restrict__ W,
    float* __restrict__ xp, int N) {
  __shared__ float xs[128 * LDS_STRIDE];
  const int tid  = threadIdx.x;
  const int wave = tid >> 5;
  const int lane = tid & 31;
  const int block_row0 = blockIdx.x * 128;

  // Async stage of the 128x64 fp32 x-tile into LDS: per-lane 16 B transfers,
  // memory -> LDS directly (GLOBAL_LOAD_ASYNC_TO_LDS_B128, tracked by ASYNCcnt).
  // Rows >= N are skipped: they are never read (tail waves exit uniformly).
  for (int i = tid; i < 128 * 16; i += 256) {
    const int r = i >> 4, c4 = i & 15;
    const int gr = block_row0 + r;
    if (gr < N) {
      const unsigned lds_addr = (unsigned)(unsigned long long)
          (__attribute__((address_space(3))) char*)(&xs[r * LDS_STRIDE + c4 * 4]);
      const unsigned long long gaddr =
          (unsigned long long)(x + (size_t)gr * INDIM + c4 * 4);
      asm volatile("global_load_async_to_lds_b128 %0, %1, off"
                   :: "v"(lds_addr), "v"(gaddr) : "memory");
    }
  }
#if __has_builtin(__builtin_amdgcn_s_wait_asynccnt)
  __builtin_amdgcn_s_wait_asynccnt(0);
#else
  asm volatile("s_wait_asynccnt 0x0" ::: "memory");
#endif
  __syncthreads();

  const int mrow0 = block_row0 + wave * 16;
  if (mrow0 >= N) return;                 // wave-uniform exit: EXEC stays all-1s

  const int rlo  = lane & 15;             // A row / B-D column within tile
  const int koff = (lane >> 4) * 2;       // K sub-offset per ISA 32-bit layouts

  // A fragments (16x4 fp32, ISA 7.12.2): lane holds row rlo, K = kb*4+koff+{0,1}
  v2f a[16];
#pragma unroll
  for (int kb = 0; kb < 16; ++kb) {
    const float* p = &xs[(wave * 16 + rlo) * LDS_STRIDE + kb * 4 + koff];
    a[kb][0] = p[0];
    a[kb][1] = p[1];
  }

#pragma unroll 1
  for (int nt = 0; nt < 16; ++nt) {
    const int ocol = nt * 16 + rlo;       // output column this lane covers
    // B fragments (4x16): lane holds col ocol, K = kb*4+koff+{0,1};
    // B[k][o] = W[o][k] -> consecutive in W's row (8B-aligned pair, L2-resident).
    v2f b[16];
#pragma unroll
    for (int kb = 0; kb < 16; ++kb) {
      const float* p = W + ocol * INDIM + kb * 4 + koff;
      b[kb][0] = p[0];
      b[kb][1] = p[1];
    }
    v8f acc = {0.f, 0.f, 0.f, 0.f, 0.f, 0.f, 0.f, 0.f};
#pragma unroll
    for (int kb = 0; kb < 16; ++kb) {
      acc = __builtin_amdgcn_wmma_f32_16x16x4_f32(
          /*neg_a=*/false, a[kb], /*neg_b=*/false, b[kb],
          /*c_mod=*/(short)0, acc, /*reuse_a=*/false, /*reuse_b=*/false);
    }
    // D layout: VGPR v -> row v (lanes 0-15) / v+8 (lanes 16-31), col = lane&15.
    const int rbase = mrow0 + ((lane >> 4) << 3);
#pragma unroll
    for (int v = 0; v < 8; ++v) {
      xp[(size_t)(rbase + v) * HC + ocol] = acc[v];
    }
  }
}

// ---------------------------------------------------------------------------
// Kernel 2: per-node attention logits  a_src/a_dst [N*H]
// ---------------------------------------------------------------------------
__global__ void gat_logits(const float* __restrict__ xp,
                           const float* __restrict__ att_src,
                           const float* __restrict__ att_dst,
                           float* __restrict__ a_src, float* __restrict__ a_dst,
                           int N) {
  const int i = blockIdx.x * blockDim.x + threadIdx.x;   // i = n*H + h
  if (i >= N * HEADS) return;
  const int h = i & (HEADS - 1);
  const float* row = xp + (size_t)i * CDIM;
  const float* as  = att_src + h * CDIM;
  const float* ad  = att_dst + h * CDIM;
  float s0 = 0.f, s1 = 0.f;
#pragma unroll
  for (int c = 0; c < CDIM; c += 4) {
    const float4 v = *(const float4*)(row + c);
    const float4 u = *(const float4*)(as + c);
    const float4 w = *(const float4*)(ad + c);
    s0 += v.x * u.x + v.y * u.y + v.z * u.z + v.w * u.w;
    s1 += v.x * w.x + v.y * w.y + v.z * w.z + v.w * w.w;
  }
  a_src[i] = s0;
  a_dst[i] = s1;
}

// ---------------------------------------------------------------------------
// Monotonic float <-> uint mapping so unsigned atomicMax == float max.
// f2ord(-inf)=0x007FFFFF > 0, so zero-init behaves as -inf sentinel (every
// node has a self-loop, so every segment receives at least one finite value).
// ---------------------------------------------------------------------------
__device__ __forceinline__ unsigned f2ord(float f) {
  unsigned u = __float_as_uint(f);
  return (u & 0x80000000u) ? ~u : (u | 0x80000000u);
}
__device__ __forceinline__ float ord2f(unsigned v) {
  return __uint_as_float((v & 0x80000000u) ? (v & 0x7fffffffu) : ~v);
}

// Kernel 3: zero/init scratch (graph-capture safe; no hipMemset dependence).
__global__ void gat_init(unsigned* __restrict__ maxbits, float* __restrict__ denom,
                         float* __restrict__ agg, int nmh, int nagg) {
  const int i = blockIdx.x * blockDim.x + threadIdx.x;
  if (i < nagg) agg[i] = 0.f;
  if (i < nmh) { maxbits[i] = 0u; denom[i] = 0.f; }
}

__device__ __forceinline__ void edge_endpoints(const int* __restrict__ ei,
                                               int e, int E, int& s, int& d) {
  if (e < E) { s = ei[e]; d = ei[E + e]; }   // edge_index row-major [2,E]
  else       { s = d = e - E; }              // appended self-loop
}

// Kernel 4: per-(edge,head) leaky-relu logit + segment max (atomic on mapped u32).
__global__ void gat_edge_max(const int* __restrict__ ei,
                             const float* __restrict__ a_src,
                             const float* __restrict__ a_dst,
                             float* __restrict__ e_ws,
                             unsigned* __restrict__ maxbits, int E, int Etot) {
  const int i = blockIdx.x * blockDim.x + threadIdx.x;   // i = e*H + h
  if (i >= Etot * HEADS) return;
  const int e = i >> 2, h = i & 3;
  int s, d;
  edge_endpoints(ei, e, E, s, d);
  float v = a_src[s * HEADS + h] + a_dst[d * HEADS + h];
  v = (v > 0.f) ? v : NEG_SLOPE * v;
  e_ws[i] = v;
  atomicMax(&maxbits[d * HEADS + h], f2ord(v));
}

// Kernel 5: ex = exp(e - m[dst]); denominator segment-sum.
__global__ void gat_edge_exp(const int* __restrict__ ei,
                             float* __restrict__ e_ws,
                             const unsigned* __restrict__ maxbits,
                             float* __restrict__ denom, int E, int Etot) {
  const int i = blockIdx.x * blockDim.x + threadIdx.x;
  if (i >= Etot * HEADS) return;
  const int e = i >> 2, h = i & 3;
  int s, d;
  edge_endpoints(ei, e, E, s, d);
  const float m  = ord2f(maxbits[d * HEADS + h]);
  const float ex = expf(e_ws[i] - m);
  e_ws[i] = ex;
  atomicAdd(&denom[d * HEADS + h], ex);
}

// Kernel 6 (the heavy one): one wave per edge; 256-float gather of xp[src]
// (coalesced 128 B per (head,half)) and 256 float atomicAdds into agg[dst].
__global__ __launch_bounds__(256) void gat_scatter(
    const int* __restrict__ ei, const float* __restrict__ xp,
    const float* __restrict__ e_ws, const float* __restrict__ denom,
    float* __restrict__ agg, int E, int Etot) {
  const int wid  = (blockIdx.x * blockDim.x + threadIdx.x) >> 5;
  const int lane = threadIdx.x & 31;
  if (wid >= Etot) return;
  int s, d;
  edge_endpoints(ei, wid, E, s, d);
  const size_t sb = (size_t)s * HC, db = (size_t)d * HC;
#pragma unroll
  for (int h = 0; h < HEADS; ++h) {
    const float alpha = e_ws[wid * HEADS + h] / (denom[d * HEADS + h] + 1e-16f);
    const int off = h * CDIM + lane;
    atomicAdd(&agg[db + off],      alpha * xp[sb + off]);
    atomicAdd(&agg[db + off + 32], alpha * xp[sb + off + 32]);
  }
}

// Kernel 7: one wave per node: mean over heads + bias, relu, LayerNorm
// (wave-shuffle reduction over 64 channels), residual add.
__global__ __launch_bounds__(256) void gat_finalize(
    const float* __restrict__ agg, const float* __restrict__ x,
    const float* __restrict__ bias, const float* __restrict__ gamma,
    const float* __restrict__ beta, float* __restrict__ out, int N) {
  const int wid  = (blockIdx.x * blockDim.x + threadIdx.x) >> 5;
  const int lane = threadIdx.x & 31;
  if (wid >= N) return;
  const size_t nb = (size_t)wid * HC;
  float t[2];
  float lsum = 0.f, lsq = 0.f;
#pragma unroll
  for (int j = 0; j < 2; ++j) {
    const int c = lane + j * 32;
    float s = 0.f;
#pragma unroll
    for (int h = 0; h < HEADS; ++h) s += agg[nb + h * CDIM + c];
    float v = s * (1.f / HEADS) + bias[c];
    v = fmaxf(v, 0.f);                      // relu
    t[j] = v;
    lsum += v;
    lsq  += v * v;
  }
#pragma unroll
  for (int m = 16; m >= 1; m >>= 1) {       // wave32 butterfly reduction
    lsum += __shfl_xor(lsum, m, 32);
    lsq  += __shfl_xor(lsq,  m, 32);
  }
  const float mu  = lsum * (1.f / CDIM);
  const float var = lsq * (1.f / CDIM) - mu * mu;   // population variance
  const float inv = rsqrtf(var + LN_EPS);
#pragma unroll
  for (int j = 0; j < 2; ++j) {
    const int c = lane + j * 32;
    out[(size_t)wid * CDIM + c] =
        (t[j] - mu) * inv * gamma[c] + beta[c] + x[(size_t)wid * CDIM + c];
  }
}

// ---------------------------------------------------------------------------
extern "C" void kernel_launch(void* const* d_in, const int* in_sizes, int n_in,
                              void* d_out, int out_size, void* d_ws, size_t ws_size,
                              hipStream_t stream) {
  const float* x        = (const float*)d_in[0];
  const int*   ei       = (const int*)d_in[1];   // [2, E] row-major
  const float* W        = (const float*)d_in[2]; // [256, 64]
  const float* att_src  = (const float*)d_in[3]; // [4, 64]
  const float* att_dst  = (const float*)d_in[4];
  const float* bias     = (const float*)d_in[5];
  const float* ln_gamma = (const float*)d_in[6];
  const float* ln_beta  = (const float*)d_in[7];

  const int N    = in_sizes[0] / INDIM;
  const int E    = in_sizes[1] / 2;
  const int Etot = E + N;                        // + self loops

  // Workspace carve-out (256B aligned).
  char* ws = (char*)d_ws;
  size_t off = 0;
  auto carve = [&](size_t bytes) {
    void* p = ws + off;
    off = (off + bytes + 255) & ~(size_t)255;
    return p;
  };
  float*    xp      = (float*)   carve((size_t)N * HC * sizeof(float));
  float*    agg     = (float*)   carve((size_t)N * HC * sizeof(float));
  float*    a_src   = (float*)   carve((size_t)N * HEADS * sizeof(float));
  float*    a_dst   = (float*)   carve((size_t)N * HEADS * sizeof(float));
  unsigned* maxbits = (unsigned*)carve((size_t)N * HEADS * sizeof(unsigned));
  float*    denom   = (float*)   carve((size_t)N * HEADS * sizeof(float));
  float*    e_ws    = (float*)   carve((size_t)Etot * HEADS * sizeof(float));
  (void)ws_size;

  float* out = (float*)d_out;

  // 1) fp32 WMMA projection (async-LDS staged)
  gat_gemm_wmma<<<(N + 127) / 128, 256, 0, stream>>>(x, W, xp, N);
  // 2) per-node logits
  gat_logits<<<(N * HEADS + 255) / 256, 256, 0, stream>>>(xp, att_src, att_dst,
                                                          a_src, a_dst, N);
  // 3) init accumulators
  gat_init<<<(N * HC + 255) / 256, 256, 0, stream>>>(maxbits, denom, agg,
                                                     N * HEADS, N * HC);
  // 4) edge logits + segment max
  gat_edge_max<<<(Etot * HEADS + 255) / 256, 256, 0, stream>>>(
      ei, a_src, a_dst, e_ws, maxbits, E, Etot);
  // 5) exp + segment sum
  gat_edge_exp<<<(Etot * HEADS + 255) / 256, 256, 0, stream>>>(
      ei, e_ws, maxbits, denom, E, Etot);
  // 6) weighted scatter aggregation (wave per edge)
  gat_scatter<<<(Etot * 32 + 255) / 256, 256, 0, stream>>>(
      ei, xp, e_ws, denom, agg, E, Etot);
  // 7) head-mean + bias + relu + LayerNorm + residual (wave per node)
  gat_finalize<<<(N * 32 + 255) / 256, 256, 0, stream>>>(
      agg, x, bias, ln_gamma, ln_beta, out, N);
}